// EighMSE_42880953483664
// MI455X (gfx1250) — compile-verified
//
#include <hip/hip_runtime.h>
#include <hip/hip_bf16.h>

typedef __attribute__((ext_vector_type(2))) float v2f;
typedef __attribute__((ext_vector_type(8))) float v8f;

// Closed-form eigendecomposition of [[a,b],[b,c]].
// l1 >= l2 (analytic order used by the reference).
// Eigenvector for l1 is (vc, vs); for l2 it is (-vs, vc).
// Jacobi convention: 2*theta = atan2(2b, a-c), cos(theta) >= 0.
__device__ __forceinline__ void eig2(float a, float b, float c,
                                     float& l1, float& l2,
                                     float& vc, float& vs) {
    float tr  = a + c;
    float amc = a - c;
    float d   = sqrtf(fmaf(amc, amc, 4.0f * b * b));
    l1 = 0.5f * (tr + d);
    l2 = 0.5f * (tr - d);
    if (d > 0.0f) {
        float c2 = amc / d;                                // cos(2θ)
        float cc = sqrtf(fmaxf(0.0f, 0.5f * (1.0f + c2))); // cosθ ≥ 0
        float ss = sqrtf(fmaxf(0.0f, 0.5f * (1.0f - c2))); // |sinθ|
        vc = cc;
        vs = (b >= 0.0f) ? ss : -ss;                       // sign(sin 2θ) = sign(b)
    } else {
        vc = 1.0f; vs = 0.0f;                              // degenerate (prob. 0)
    }
}

__device__ __forceinline__ void row_contrib(float pa, float pb, float pc,
                                            float ta, float tb, float tc,
                                            float& s_eval, float& s_evec) {
    float pl1, pl2, pvc, pvs, tl1, tl2, tvc, tvs;
    eig2(pa, pb, pc, pl1, pl2, pvc, pvs);
    eig2(ta, tb, tc, tl1, tl2, tvc, tvs);
    float d1 = pl1 - tl1, d2 = pl2 - tl2;
    s_eval += d1 * d1 + d2 * d2;
    float dx = pvc - tvc, dy = pvs - tvs;
    // |p2 - t2|^2 == |p1 - t1|^2 under this sign convention, so one term covers both.
    s_evec += dx * dx + dy * dy;
}

// Stage 1: per-block partial sums. 4 rows/thread so loads are 3x float4 per input
// (48B per thread per array, 16B aligned -> global_load_b128, fully coalesced).
__global__ __launch_bounds__(256)
void eigh_mse_partial(const float* __restrict__ yp, const float* __restrict__ yt,
                      float* __restrict__ ws_eval, float* __restrict__ ws_evec,
                      int B) {
    const int  t  = blockIdx.x * blockDim.x + threadIdx.x;
    const long r0 = (long)t * 4;
    float se = 0.0f, sv = 0.0f;

    if (r0 + 3 < (long)B) {
        const float4* p4 = reinterpret_cast<const float4*>(yp) + (size_t)t * 3;
        const float4* q4 = reinterpret_cast<const float4*>(yt) + (size_t)t * 3;
        float4 p0 = p4[0], p1 = p4[1], p2 = p4[2];
        float4 q0 = q4[0], q1 = q4[1], q2 = q4[2];
        row_contrib(p0.x, p0.y, p0.z, q0.x, q0.y, q0.z, se, sv);
        row_contrib(p0.w, p1.x, p1.y, q0.w, q1.x, q1.y, se, sv);
        row_contrib(p1.z, p1.w, p2.x, q1.z, q1.w, q2.x, se, sv);
        row_contrib(p2.y, p2.z, p2.w, q2.y, q2.z, q2.w, se, sv);
    } else {
        for (long r = r0; r < (long)B && r < r0 + 4; ++r) {
            row_contrib(yp[r * 3], yp[r * 3 + 1], yp[r * 3 + 2],
                        yt[r * 3], yt[r * 3 + 1], yt[r * 3 + 2], se, sv);
        }
    }

    // wave32 tree reduction
    for (int m = 16; m > 0; m >>= 1) {
        se += __shfl_xor(se, m, 32);
        sv += __shfl_xor(sv, m, 32);
    }

    __shared__ float sm[2][8];
    const int wave = threadIdx.x >> 5;
    const int lane = threadIdx.x & 31;
    if (lane == 0) { sm[0][wave] = se; sm[1][wave] = sv; }
    __syncthreads();
    if (threadIdx.x == 0) {
        const int nw = blockDim.x >> 5;
        float a = 0.0f, b = 0.0f;
        for (int i = 0; i < nw; ++i) { a += sm[0][i]; b += sm[1][i]; }
        ws_eval[blockIdx.x] = a;
        ws_evec[blockIdx.x] = b;
    }
}

// Zero the pad region [n, npad) of both partial arrays so stage 2 can load
// unguarded. Skipped entirely when n is already a multiple of 64.
__global__ __launch_bounds__(64)
void eigh_mse_zeropad(float* __restrict__ ws_eval, float* __restrict__ ws_evec,
                      int n, int npad) {
    int i = n + threadIdx.x;
    if (i < npad) { ws_eval[i] = 0.0f; ws_evec[i] = 0.0f; }
}

// Stage 2: one wave32. WMMA F32 16x16x4 with B = ones acts as a 64-wide fp32
// accumulator: D[i][j] = sum_k A[i][k] + C[i][j]. After folding all chunks,
// sum(C) == 16 * grand_total (each row-sum replicated across 16 columns).
// nparts_pad is a multiple of 64 and the pad entries are zero, so each chunk
// is one unguarded aligned global_load_b64 per lane (any lane<->A bijection
// sums identically).
__global__ __launch_bounds__(32)
void eigh_mse_final(const float* __restrict__ ws_eval,
                    const float* __restrict__ ws_evec,
                    const float* __restrict__ weights,
                    float* __restrict__ out,
                    int nparts_pad, int B) {
    const int lane = threadIdx.x;
    v2f ones; ones.x = 1.0f; ones.y = 1.0f;

    float totals[2];
    const float* srcs[2] = { ws_eval, ws_evec };
    const int nchunks = nparts_pad >> 6;

    for (int a = 0; a < 2; ++a) {
        const v2f* p = reinterpret_cast<const v2f*>(srcs[a]);  // 8B-aligned pairs
        v8f c = {};
        for (int ch = 0; ch < nchunks; ++ch) {
            v2f av = p[ch * 32 + lane];   // global_load_b64, fully coalesced
            // D = A(16x4) x ones(4x16) + C : accumulates 64 values into C tile
            c = __builtin_amdgcn_wmma_f32_16x16x4_f32(false, av, false, ones,
                                                      (short)0, c, false, false);
        }
        float s = c[0] + c[1] + c[2] + c[3] + c[4] + c[5] + c[6] + c[7];
        for (int m = 16; m > 0; m >>= 1) s += __shfl_xor(s, m, 32);
        totals[a] = s * 0.0625f;   // /16 is exact (row-sums replicated x16)
    }

    if (lane == 0) {
        const float w0 = weights[0], w1 = weights[1], w2 = weights[2];
        const float invB = 1.0f / (float)B;
        // w0 * mean over (B,2) of dλ²  +  (w1+w2)/2 * (Σ|Δv|²) / B
        out[0] = w0 * totals[0] * (0.5f * invB)
               + 0.5f * (w1 + w2) * totals[1] * invB;
    }
}

extern "C" void kernel_launch(void* const* d_in, const int* in_sizes, int n_in,
                              void* d_out, int out_size, void* d_ws, size_t ws_size,
                              hipStream_t stream) {
    const int B = in_sizes[0] / 3;
    const float* yp = (const float*)d_in[0];
    const float* yt = (const float*)d_in[1];
    const float* w  = (const float*)d_in[2];
    float* out = (float*)d_out;

    const int nthreads = (B + 3) / 4;
    const int blocks   = (nthreads + 255) / 256;   // 2048 for B = 2^21
    const int pad      = (blocks + 63) & ~63;      // multiple of 64 for stage 2

    float* ws_eval = (float*)d_ws;
    float* ws_evec = ws_eval + pad;                // keeps 8B (in fact 256B) alignment

    eigh_mse_partial<<<blocks, 256, 0, stream>>>(yp, yt, ws_eval, ws_evec, B);
    if (pad > blocks) {
        eigh_mse_zeropad<<<1, 64, 0, stream>>>(ws_eval, ws_evec, blocks, pad);
    }
    eigh_mse_final<<<1, 32, 0, stream>>>(ws_eval, ws_evec, w, out, pad, B);
}